// SAGEModule_10359461118095
// MI455X (gfx1250) — compile-verified
//
#include <hip/hip_runtime.h>

#define N_NODES 100000
#define N_EDGES 640000
#define D 128
#define EPS 1e-5f
#define LDSTRIDE 132  // 128 + 4 floats pad -> bank-conflict-free column reads

typedef __attribute__((ext_vector_type(2))) float v2f;
typedef __attribute__((ext_vector_type(8))) float v8f;

// ---------------------------------------------------------------- zero scratch
__global__ void zero_ws_kernel(float* __restrict__ p, long n) {
    long i = (long)blockIdx.x * blockDim.x + threadIdx.x;
    long stride = (long)gridDim.x * blockDim.x;
    for (; i < n; i += stride) p[i] = 0.0f;
}

// ---------------------------------------------------------------- LayerNorm
// One wave32 per node: 128 features = 4 floats/lane, shfl_xor reductions.
__global__ __launch_bounds__(256) void layernorm_kernel(
    const float* __restrict__ h, const float* __restrict__ gamma,
    const float* __restrict__ beta, float* __restrict__ hn) {
    int wid  = (blockIdx.x * blockDim.x + threadIdx.x) >> 5;
    int lane = threadIdx.x & 31;
    if (wid >= N_NODES) return;

    float4 v = ((const float4*)(h + (long)wid * D))[lane];
    float s = v.x + v.y + v.z + v.w;
#pragma unroll
    for (int off = 16; off; off >>= 1) s += __shfl_xor(s, off, 32);
    float mu = s * (1.0f / D);

    float dx = v.x - mu, dy = v.y - mu, dz = v.z - mu, dw = v.w - mu;
    float q = dx * dx + dy * dy + dz * dz + dw * dw;
#pragma unroll
    for (int off = 16; off; off >>= 1) q += __shfl_xor(q, off, 32);
    float rstd = rsqrtf(q * (1.0f / D) + EPS);

    float4 g  = ((const float4*)gamma)[lane];
    float4 bt = ((const float4*)beta)[lane];
    float4 o;
    o.x = dx * rstd * g.x + bt.x;
    o.y = dy * rstd * g.y + bt.y;
    o.z = dz * rstd * g.z + bt.z;
    o.w = dw * rstd * g.w + bt.w;
    ((float4*)(hn + (long)wid * D))[lane] = o;
}

// ---------------------------------------------------------------- edge scatter
// One wave32 per edge: atomic-add hn[src] into msg[dst] (lands in L2 — the
// 51 MB feature table fits the 192 MB global L2), lane 0 bumps degree.
__global__ __launch_bounds__(256) void scatter_kernel(
    const float* __restrict__ hn, const int* __restrict__ src,
    const int* __restrict__ dst, float* __restrict__ msg,
    float* __restrict__ deg) {
    int wid  = (blockIdx.x * blockDim.x + threadIdx.x) >> 5;
    int lane = threadIdx.x & 31;
    if (wid >= N_EDGES) return;
    int s = src[wid], d = dst[wid];
    const float* hs = hn + (long)s * D;
    float* md = msg + (long)d * D;
#pragma unroll
    for (int i = 0; i < 4; ++i)
        atomicAdd(md + lane + 32 * i, hs[lane + 32 * i]);
    if (lane == 0) atomicAdd(deg + d, 1.0f);
}

// ---------------------------------------------------------------- fused GEMM
// out[32 rows x 128 cols per block] = hn@W_self + (msg/deg)@W_neigh + b
// 8 waves/block; each wave owns a 32x16 output slab as TWO 16x16 WMMA tiles
// sharing one B fragment -> halves B-operand traffic, doubles WMMA density.
__global__ __launch_bounds__(256) void sage_gemm_kernel(
    const float* __restrict__ hn, const float* __restrict__ msg,
    const float* __restrict__ deg, const float* __restrict__ Wself,
    const float* __restrict__ Wneigh, const float* __restrict__ bias,
    float* __restrict__ out) {
    __shared__ float As[32 * LDSTRIDE];
    __shared__ float An[32 * LDSTRIDE];
    __shared__ float invdeg[32];

    int t = threadIdx.x;
    int rowBase = blockIdx.x * 32;

    if (t < 32) invdeg[t] = 1.0f / fmaxf(deg[rowBase + t], 1.0f);
#pragma unroll
    for (int i = 0; i < 16; ++i) {
        int e = t + i * 256;
        int r = e >> 7, c = e & 127;
        As[r * LDSTRIDE + c] = hn[(long)(rowBase + r) * D + c];
    }
    __syncthreads();
#pragma unroll
    for (int i = 0; i < 16; ++i) {
        int e = t + i * 256;
        int r = e >> 7, c = e & 127;
        An[r * LDSTRIDE + c] = msg[(long)(rowBase + r) * D + c] * invdeg[r];
    }
    __syncthreads();

    int lane = t & 31;
    int wave = t >> 5;        // 0..7 -> which 16-col slab
    int nq   = lane & 15;     // col within tile / A row
    int hi   = lane >> 4;     // half-wave select (K pair / M+8)
    int ncol = wave * 16 + nq;
    int m    = nq;

    v8f acc0 = {};            // rows rowBase +  0..15
    v8f acc1 = {};            // rows rowBase + 16..31

    // Self part: acc += hn_tile @ W_self   (B fragment shared by both tiles)
#pragma unroll 4
    for (int k = 0; k < D; k += 4) {
        int kk = k + (hi << 1);                   // lanes 16-31 carry K=+2,+3
        v2f a0, a1, b;
        a0.x = As[m * LDSTRIDE + kk];             // A 16x4 f32: V0=K0|K2, V1=K1|K3
        a0.y = As[m * LDSTRIDE + kk + 1];
        a1.x = As[(m + 16) * LDSTRIDE + kk];
        a1.y = As[(m + 16) * LDSTRIDE + kk + 1];
        b.x  = Wself[kk * D + ncol];              // B 4x16 f32, N striped per lane
        b.y  = Wself[(kk + 1) * D + ncol];
        acc0 = __builtin_amdgcn_wmma_f32_16x16x4_f32(
            false, a0, false, b, (short)0, acc0, false, false);
        acc1 = __builtin_amdgcn_wmma_f32_16x16x4_f32(
            false, a1, false, b, (short)0, acc1, false, false);
    }
    // Neighbor part: acc += (msg/deg)_tile @ W_neigh
#pragma unroll 4
    for (int k = 0; k < D; k += 4) {
        int kk = k + (hi << 1);
        v2f a0, a1, b;
        a0.x = An[m * LDSTRIDE + kk];
        a0.y = An[m * LDSTRIDE + kk + 1];
        a1.x = An[(m + 16) * LDSTRIDE + kk];
        a1.y = An[(m + 16) * LDSTRIDE + kk + 1];
        b.x  = Wneigh[kk * D + ncol];
        b.y  = Wneigh[(kk + 1) * D + ncol];
        acc0 = __builtin_amdgcn_wmma_f32_16x16x4_f32(
            false, a0, false, b, (short)0, acc0, false, false);
        acc1 = __builtin_amdgcn_wmma_f32_16x16x4_f32(
            false, a1, false, b, (short)0, acc1, false, false);
    }

    // C/D layout: VGPR i = row M=i (lanes 0-15) / M=i+8 (lanes 16-31)
    float bv = bias[ncol];
    int mBase = hi * 8;
#pragma unroll
    for (int i = 0; i < 8; ++i) {
        out[(long)(rowBase + mBase + i) * D + ncol]      = acc0[i] + bv;
        out[(long)(rowBase + 16 + mBase + i) * D + ncol] = acc1[i] + bv;
    }
}

// ---------------------------------------------------------------- launcher
extern "C" void kernel_launch(void* const* d_in, const int* in_sizes, int n_in,
                              void* d_out, int out_size, void* d_ws, size_t ws_size,
                              hipStream_t stream) {
    const float* h      = (const float*)d_in[0];
    const int*   src    = (const int*)d_in[1];
    const int*   dst    = (const int*)d_in[2];
    const float* gamma  = (const float*)d_in[3];
    const float* beta   = (const float*)d_in[4];
    const float* Wself  = (const float*)d_in[5];
    const float* Wneigh = (const float*)d_in[6];
    const float* bias   = (const float*)d_in[7];
    float* out = (float*)d_out;

    float* hn  = (float*)d_ws;                      // [N_NODES, D]
    float* msg = hn + (long)N_NODES * D;            // [N_NODES, D]
    float* deg = msg + (long)N_NODES * D;           // [N_NODES]

    long zcount = (long)N_NODES * D + N_NODES;      // msg + deg contiguous
    zero_ws_kernel<<<2048, 256, 0, stream>>>(msg, zcount);

    layernorm_kernel<<<(N_NODES + 7) / 8, 256, 0, stream>>>(h, gamma, beta, hn);

    scatter_kernel<<<(N_EDGES + 7) / 8, 256, 0, stream>>>(hn, src, dst, msg, deg);

    sage_gemm_kernel<<<N_NODES / 32, 256, 0, stream>>>(
        hn, msg, deg, Wself, Wneigh, bias, out);
}